// HingeLoss_67345087201366
// MI455X (gfx1250) — compile-verified
//
#include <hip/hip_runtime.h>

// Multiclass hinge loss, MI455X (gfx1250, wave32).
// loss = sum_i sum_{j != y_i} max(0, 1 - (o[i,y_i] - o[i,j])) / N
//
// Memory-bound: 200 MB fp32 streamed once -> ~8.6 us floor at 23.3 TB/s.
// Per-row class reduction is done on the matrix pipe with
// V_WMMA_F32_16X16X4_F32 against an all-ones B matrix (ones-matrix
// reduction), freeing the VALU for the max/add margin math.

typedef __attribute__((ext_vector_type(2))) float v2f;
typedef __attribute__((ext_vector_type(8))) float v8f;

#define NUM_CLASSES 100
#define WAVES_PER_BLOCK 8

__global__ __launch_bounds__(256) void hinge_partial_kernel(
    const float* __restrict__ out, const int* __restrict__ labels,
    double* __restrict__ partials, int n) {
  const int lane = threadIdx.x & 31;
  const int wave = threadIdx.x >> 5;
  const int waveId = blockIdx.x * WAVES_PER_BLOCK + wave;
  const int base = waveId * 16;  // 16 rows per wave-tile

  // A-matrix (16x4 f32) lane mapping: lane holds row = lane%16,
  // K-pair (0,1) for lanes 0-15, (2,3) for lanes 16-31.
  const int r = lane & 15;
  const int hi = lane >> 4;  // 0 or 1

  const int row = base + r;
  const bool valid = row < n;
  const int rowC = valid ? row : (n - 1);  // clamp: keep loads in-bounds, EXEC stays full

  const float* rowPtr = out + (size_t)rowC * NUM_CLASSES;

  // Gather the correct-class score for this row (2 lanes/row redundantly; L0 absorbs it).
  const int lbl = labels[rowC];
  const float corr = rowPtr[lbl];
  const float bias = 1.0f - corr;

  // Hint the streamer: pull the tail of this row toward the caches.
  __builtin_prefetch(rowPtr + 64, 0, 1);

  v8f acc = {};                 // f32 C/D accumulator (16x16)
  v2f bones; bones[0] = 1.0f; bones[1] = 1.0f;  // all-ones B (4x16) -> row-sum reduction
  const float vmask = valid ? 1.0f : 0.0f;

#pragma unroll
  for (int c = 0; c < NUM_CLASSES / 4; ++c) {
    const int col = 4 * c + 2 * hi;
    // 8-byte aligned: row stride 400B, col even -> global_load_b64
    v2f v = *(const v2f*)(rowPtr + col);
    float m0 = bias + v[0]; m0 = m0 > 0.0f ? m0 : 0.0f;
    float m1 = bias + v[1]; m1 = m1 > 0.0f ? m1 : 0.0f;
    v2f a; a[0] = m0 * vmask; a[1] = m1 * vmask;
    // D = A x ones + C : accumulates per-row margin sums into every column of D
    acc = __builtin_amdgcn_wmma_f32_16x16x4_f32(
        /*neg_a=*/false, a, /*neg_b=*/false, bones,
        /*c_mod=*/(short)0, acc, /*reuse_a=*/false, /*reuse_b=*/false);
  }

  // D[m][n] == rowsum[m] for every n. Lane sum over its 8 C VGPRs gives:
  //   lanes 0-15 : sum_{m=0..7}  rowsum[m]   (identical across those lanes)
  //   lanes 16-31: sum_{m=8..15} rowsum[m]
  float laneSum = acc[0] + acc[1] + acc[2] + acc[3] +
                  acc[4] + acc[5] + acc[6] + acc[7];
  const float sLow  = __shfl(laneSum, 0, 32);
  const float sHigh = __shfl(laneSum, 16, 32);

  // j == y_i contributes exactly max(0, 1-0) = 1 per valid row: subtract count.
  int vcount = n - base;
  vcount = vcount < 0 ? 0 : (vcount > 16 ? 16 : vcount);
  const double waveTotal = (double)sLow + (double)sHigh - (double)vcount;

  __shared__ double ws[WAVES_PER_BLOCK];
  if (lane == 0) ws[wave] = waveTotal;
  __syncthreads();
  if (threadIdx.x == 0) {
    double t = 0.0;
#pragma unroll
    for (int i = 0; i < WAVES_PER_BLOCK; ++i) t += ws[i];
    partials[blockIdx.x] = t;  // deterministic fixed-order partial
  }
}

__global__ __launch_bounds__(256) void hinge_final_kernel(
    const double* __restrict__ partials, float* __restrict__ outv,
    int nblocks, double invN) {
  __shared__ double sm[256];
  double t = 0.0;
  for (int i = threadIdx.x; i < nblocks; i += 256) t += partials[i];
  sm[threadIdx.x] = t;
  __syncthreads();
  for (int s = 128; s > 0; s >>= 1) {
    if ((int)threadIdx.x < s) sm[threadIdx.x] += sm[threadIdx.x + s];
    __syncthreads();
  }
  if (threadIdx.x == 0) outv[0] = (float)(sm[0] * invN);
}

extern "C" void kernel_launch(void* const* d_in, const int* in_sizes, int n_in,
                              void* d_out, int out_size, void* d_ws, size_t ws_size,
                              hipStream_t stream) {
  const float* output = (const float*)d_in[0];
  const int* labels = (const int*)d_in[1];
  const int n = in_sizes[1];  // number of rows (label count)

  const int nTiles = (n + 15) / 16;                              // 16 rows per wave
  const int nBlocks = (nTiles + WAVES_PER_BLOCK - 1) / WAVES_PER_BLOCK;

  double* partials = (double*)d_ws;

  hinge_partial_kernel<<<nBlocks, 256, 0, stream>>>(output, labels, partials, n);
  hinge_final_kernel<<<1, 256, 0, stream>>>(partials, (float*)d_out, nBlocks,
                                            1.0 / (double)n);
}